// KNN_InstanceLoss_78331613544660
// MI455X (gfx1250) — compile-verified
//
#include <hip/hip_runtime.h>
#include <hip/hip_bf16.h>
#include <stdint.h>
#include <math.h>

// ---------------------------------------------------------------------------
// KNN instance loss, MI455X (gfx1250).
//   B=4096 rows, D=1024 feature dim, K=16 clusters, T=0.5, thresh=0.5
//
// Stages (all on `stream`, scratch in d_ws):
//   K1  pos_mask bits (4096x128 u32) + per-row popcount
//   K2  pos_min = min(count), neg_min = B - max(count); zero loss accum
//   K3  cos_sim = z_i @ z_j^T via V_WMMA_F32_16X16X4_F32 (fp32 matrix pipe)
//   K4  per-row bitonic sort (neg then pos) in LDS -> fused logsumexp loss
// ---------------------------------------------------------------------------

typedef __attribute__((ext_vector_type(2))) float v2f;
typedef __attribute__((ext_vector_type(8))) float v8f;

#define BN   4096   // batch
#define DK   1024   // feature dim
#define KC   16     // clusters
#define LDT  36     // padded LDS tile row stride in floats (16B-aligned rows)

// ---- K1: pos mask + per-row pos count -------------------------------------
__global__ void __launch_bounds__(128) k_mask_count(
    const float* __restrict__ ci, const float* __restrict__ cj,
    uint32_t* __restrict__ mask, int* __restrict__ poscnt)
{
  __shared__ float sci[KC], scj[KC];
  __shared__ int scnt[128];
  const int row = blockIdx.x, t = threadIdx.x;
  if (t < KC) { sci[t] = ci[row*KC + t]; scj[t] = cj[row*KC + t]; }
  __syncthreads();
  uint32_t bits = 0; int cnt = 0;
  const int col0 = t * 32;
  for (int j = 0; j < 32; ++j) {
    const int col = col0 + j;
    float d = 0.f;
#pragma unroll
    for (int k = 0; k < KC; ++k)
      d += sci[k]*ci[col*KC + k] + scj[k]*cj[col*KC + k];
    float lm = 0.5f * d;
    if (col == row) lm = 1.0f;
    const bool p = lm > 0.5f;
    bits |= (uint32_t)p << j;
    cnt += (int)p;
  }
  mask[row*128 + t] = bits;
  scnt[t] = cnt; __syncthreads();
  for (int s = 64; s > 0; s >>= 1) { if (t < s) scnt[t] += scnt[t+s]; __syncthreads(); }
  if (t == 0) poscnt[row] = scnt[0];
}

// ---- K2: pos_min / neg_min + zero the loss accumulator --------------------
__global__ void __launch_bounds__(256) k_minmax(
    const int* __restrict__ poscnt, int* __restrict__ scal, float* __restrict__ out)
{
  __shared__ int smin[256], smax[256];
  const int t = threadIdx.x;
  int mn = 0x7fffffff, mx = 0;
  for (int i = t; i < BN; i += 256) {
    const int v = poscnt[i];
    mn = min(mn, v); mx = max(mx, v);
  }
  smin[t] = mn; smax[t] = mx; __syncthreads();
  for (int s = 128; s > 0; s >>= 1) {
    if (t < s) { smin[t] = min(smin[t], smin[t+s]); smax[t] = max(smax[t], smax[t+s]); }
    __syncthreads();
  }
  if (t == 0) { scal[0] = smin[0]; scal[1] = BN - smax[0]; out[0] = 0.f; }
}

// ---- K3: cos_sim = z_i @ z_j^T, fp32 WMMA ---------------------------------
// 128x128 C tile per 256-thread block; 8 wave32 in a 4(M) x 2(N) grid;
// each wave owns a 32x64 sub-tile = 2x4 accumulators of v8f.
__global__ void __launch_bounds__(256) k_gemm_wmma(
    const float* __restrict__ zi, const float* __restrict__ zj,
    float* __restrict__ cosm)
{
  __shared__ float Asm[128 * LDT];
  __shared__ float Bsm[128 * LDT];
  const int tileM = blockIdx.x * 128;
  const int tileN = blockIdx.y * 128;
  const int tid  = threadIdx.x;
  const int wave = tid >> 5;
  const int lane = tid & 31;
  const int wm = (wave & 3) * 32;   // wave M offset within tile
  const int wn = (wave >> 2) * 64;  // wave N offset within tile

  v8f acc[2][4];
#pragma unroll
  for (int mi = 0; mi < 2; ++mi)
#pragma unroll
    for (int ni = 0; ni < 4; ++ni)
#pragma unroll
      for (int g = 0; g < 8; ++g) acc[mi][ni][g] = 0.f;

  const int lrow  = lane & 15;         // M (A) / N (B) index within fragment
  const int khalf = (lane >> 4) * 2;   // lanes 16-31 carry K=2,3 (ISA layout)

  for (int kc = 0; kc < DK; kc += 32) {
    // stage A (z_i rows) and B (z_j rows) 128x32 f32 tiles, float4 fills
#pragma unroll
    for (int i = 0; i < 4; ++i) {
      const int lin = tid + i*256;       // 1024 float4 slots per tile
      const int r = lin >> 3, c4 = lin & 7;
      const float4 a = *(const float4*)(zi + (size_t)(tileM + r)*DK + kc + c4*4);
      *(float4*)(&Asm[r*LDT + c4*4]) = a;
      const float4 b = *(const float4*)(zj + (size_t)(tileN + r)*DK + kc + c4*4);
      *(float4*)(&Bsm[r*LDT + c4*4]) = b;
    }
    __syncthreads();

#pragma unroll
    for (int k0 = 0; k0 < 32; k0 += 4) {
      v2f afrag[2], bfrag[4];
#pragma unroll
      for (int mi = 0; mi < 2; ++mi)
        afrag[mi] = *(const v2f*)(&Asm[(wm + mi*16 + lrow)*LDT + k0 + khalf]);
#pragma unroll
      for (int ni = 0; ni < 4; ++ni)
        bfrag[ni] = *(const v2f*)(&Bsm[(wn + ni*16 + lrow)*LDT + k0 + khalf]);
#pragma unroll
      for (int mi = 0; mi < 2; ++mi)
#pragma unroll
        for (int ni = 0; ni < 4; ++ni)
          acc[mi][ni] = __builtin_amdgcn_wmma_f32_16x16x4_f32(
              false, afrag[mi], false, bfrag[ni],
              (short)0, acc[mi][ni], false, false);
    }
    __syncthreads();
  }

  // C layout: VGPR g, lane L -> row = (L<16 ? g : g+8), col = L&15
  const int rhalf = (lane >> 4) * 8;
#pragma unroll
  for (int mi = 0; mi < 2; ++mi)
#pragma unroll
    for (int ni = 0; ni < 4; ++ni)
#pragma unroll
      for (int g = 0; g < 8; ++g) {
        const int r = tileM + wm + mi*16 + rhalf + g;
        const int c = tileN + wn + ni*16 + (lane & 15);
        cosm[(size_t)r*BN + c] = acc[mi][ni][g];
      }
}

// ---- bitonic sort, descending, n=4096, 256 threads ------------------------
__device__ inline void bitonic_sort_desc_4096(float* __restrict__ v, int t)
{
  for (int k = 2; k <= 4096; k <<= 1) {
    for (int j = k >> 1; j > 0; j >>= 1) {
      for (int i = t; i < 4096; i += 256) {
        const int ixj = i ^ j;
        if (ixj > i) {
          const float a = v[i], b = v[ixj];
          const bool desc = ((i & k) == 0);
          if (desc ? (a < b) : (a > b)) { v[i] = b; v[ixj] = a; }
        }
      }
      __syncthreads();
    }
  }
}

// ---- K4: per-row top-k + fused CE loss ------------------------------------
__global__ void __launch_bounds__(256) k_topk_loss(
    const float* __restrict__ cosm, const uint32_t* __restrict__ mask,
    const int* __restrict__ scal, float* __restrict__ out)
{
  __shared__ float v[4096];
  __shared__ float red[256];
  __shared__ float Su_s, mneg_s;
  const int row = blockIdx.x, t = threadIdx.x;
  const int pos_min = scal[0];
  const int neg_min = scal[1];
  const float* crow = cosm + (size_t)row * BN;
  const uint32_t* mrow = mask + row * 128;
  const float NEGINF = -__builtin_inff();
  const float invT = 2.0f;   // 1 / 0.5

  // ---- negatives: sort (pos -> -inf), logsumexp over first neg_min ----
  for (int i = t; i < 4096; i += 256) {
    const bool p = (mrow[i >> 5] >> (i & 31)) & 1u;
    v[i] = p ? NEGINF : crow[i];
  }
  __syncthreads();
  bitonic_sort_desc_4096(v, t);
  const float mneg = v[0];
  float s = 0.f;
  for (int i = t; i < neg_min; i += 256) s += expf((v[i] - mneg) * invT);
  red[t] = s; __syncthreads();
  for (int st = 128; st > 0; st >>= 1) { if (t < st) red[t] += red[t+st]; __syncthreads(); }
  if (t == 0) { Su_s = red[0]; mneg_s = mneg; }
  __syncthreads();

  // ---- positives: sort (neg -> -inf), accumulate lse - pos/T ----
  for (int i = t; i < 4096; i += 256) {
    const bool p = (mrow[i >> 5] >> (i & 31)) & 1u;
    v[i] = p ? crow[i] : NEGINF;
  }
  __syncthreads();
  bitonic_sort_desc_4096(v, t);

  const float mu = mneg_s * invT;
  const float Su = Su_s;
  float accum = 0.f;
  for (int p = t; p < pos_min; p += 256) {
    const float a = v[p] * invT;
    const float big = fmaxf(a, mu);
    const float lse = big + logf(expf(a - big) + Su * expf(mu - big));
    accum += lse - a;
  }
  red[t] = accum; __syncthreads();
  for (int st = 128; st > 0; st >>= 1) { if (t < st) red[t] += red[t+st]; __syncthreads(); }
  if (t == 0)
    atomicAdd(out, red[0] / ((float)BN * (float)pos_min));
}

// ---------------------------------------------------------------------------
extern "C" void kernel_launch(void* const* d_in, const int* in_sizes, int n_in,
                              void* d_out, int out_size, void* d_ws, size_t ws_size,
                              hipStream_t stream) {
  const float* zi = (const float*)d_in[0];   // [4096,1024]
  const float* zj = (const float*)d_in[1];   // [4096,1024]
  const float* ci = (const float*)d_in[2];   // [4096,16]
  const float* cj = (const float*)d_in[3];   // [4096,16]
  float* out = (float*)d_out;                // scalar loss

  char* ws = (char*)d_ws;
  float*    cosm   = (float*)ws;                                       // 64 MB
  uint32_t* mask   = (uint32_t*)(ws + (size_t)BN*BN*4);                // 2 MB
  int*      poscnt = (int*)(ws + (size_t)BN*BN*4 + (size_t)BN*128*4);  // 16 KB
  int*      scal   = poscnt + BN;                                      // [pos_min, neg_min]

  k_mask_count<<<BN, 128, 0, stream>>>(ci, cj, mask, poscnt);
  k_minmax<<<1, 256, 0, stream>>>(poscnt, scal, out);
  dim3 grid(BN/128, BN/128);
  k_gemm_wmma<<<grid, 256, 0, stream>>>(zi, zj, cosm);
  k_topk_loss<<<BN, 256, 0, stream>>>(cosm, mask, scal, out);
}